// SyntheticTripletLoss_77464030151303
// MI455X (gfx1250) — compile-verified
//
#include <hip/hip_runtime.h>
#include <stdint.h>

typedef __attribute__((ext_vector_type(2))) float v2f;
typedef __attribute__((ext_vector_type(8))) float v8f;

namespace {
constexpr int   kN       = 131072;   // rows
constexpr int   kD       = 512;      // cols
constexpr int   kTPB     = 128;      // 4 waves per block (wave32)
constexpr int   kWaves   = kTPB / 32;
constexpr int   kBlocks  = 2048;     // 8192 waves total -> 16 rows/wave
constexpr float kMargin  = 0.5f;
constexpr float kEps     = 1e-12f;
}

// Copy one 512-float row (2 KB) from global memory into LDS with the CDNA5
// async-DMA path. Each lane moves 16 B per instruction; 4 instructions cover
// the row, each a fully coalesced 512 B burst. Tracked by ASYNCcnt.
__device__ __forceinline__ void async_row_to_lds(const float* __restrict__ gsrc,
                                                 uint32_t lds_base_bytes,
                                                 int lane) {
#pragma unroll
  for (int s = 0; s < 4; ++s) {
    const float* g = gsrc + s * 128 + lane * 4;
    uint32_t     l = lds_base_bytes + (uint32_t)(s * 512 + lane * 16);
    asm volatile("global_load_async_to_lds_b128 %0, %1, off"
                 :: "v"(l), "v"(g)
                 : "memory");
  }
}

__device__ __forceinline__ void wait_async_le8()  { asm volatile("s_wait_asynccnt 0x8" ::: "memory"); }
__device__ __forceinline__ void wait_async_zero() { asm volatile("s_wait_asynccnt 0x0" ::: "memory"); }
__device__ __forceinline__ void wait_ds_zero()    { asm volatile("s_wait_dscnt 0x0"    ::: "memory"); }

__global__ __launch_bounds__(kTPB) void triplet_loss_main(
    const float* __restrict__ targets,
    const float* __restrict__ preds,
    float* __restrict__ partial) {
  // [stage][wave][ preds(512) | targets(512) ]  -> 32 KB staging
  __shared__ __align__(16) float smem[2][kWaves][2 * kD];
  __shared__ float wsum[kWaves];

  const int lane   = threadIdx.x & 31;
  const int wave   = threadIdx.x >> 5;
  const int wgid   = blockIdx.x * kWaves + wave;   // global wave id
  const int nwaves = gridDim.x * kWaves;

  float lsum = 0.0f;
  int   stage = 0;
  int   r = wgid;

  if (r < kN) {
    // Low 32 bits of the flat shared address == LDS byte offset (ISA 10.2).
    uint32_t b = (uint32_t)(uintptr_t)&smem[0][wave][0];
    async_row_to_lds(preds   + (size_t)r * kD, b,              lane);
    async_row_to_lds(targets + (size_t)r * kD, b + kD * 4u,    lane);
  }

  for (; r < kN; r += nwaves) {
    const int nxt = r + nwaves;
    if (nxt < kN) {
      wait_ds_zero();  // stage^1 reads from the previous iteration fully drained
      uint32_t b = (uint32_t)(uintptr_t)&smem[stage ^ 1][wave][0];
      async_row_to_lds(preds   + (size_t)nxt * kD, b,           lane);
      async_row_to_lds(targets + (size_t)nxt * kD, b + kD * 4u, lane);
      wait_async_le8();   // 8 outstanding = the prefetch we just issued
    } else {
      wait_async_zero();
    }

    const float* bp = &smem[stage][wave][0];
    const float* bt = &smem[stage][wave][kD];

    float pp = 0.0f, pt = 0.0f, tt = 0.0f;
#pragma unroll
    for (int s = 0; s < 4; ++s) {
      const float4 p = *reinterpret_cast<const float4*>(bp + s * 128 + lane * 4);
      const float4 t = *reinterpret_cast<const float4*>(bt + s * 128 + lane * 4);
      pp = fmaf(p.x, p.x, pp); pp = fmaf(p.y, p.y, pp);
      pp = fmaf(p.z, p.z, pp); pp = fmaf(p.w, p.w, pp);
      pt = fmaf(p.x, t.x, pt); pt = fmaf(p.y, t.y, pt);
      pt = fmaf(p.z, t.z, pt); pt = fmaf(p.w, t.w, pt);
      tt = fmaf(t.x, t.x, tt); tt = fmaf(t.y, t.y, tt);
      tt = fmaf(t.z, t.z, tt); tt = fmaf(t.w, t.w, tt);
    }

    // wave32 butterfly reduction
#pragma unroll
    for (int off = 16; off > 0; off >>= 1) {
      pp += __shfl_xor(pp, off, 32);
      pt += __shfl_xor(pt, off, 32);
      tt += __shfl_xor(tt, off, 32);
    }

    // loss = relu(margin + (pp - s^2)/max(||p - s t||, eps) - s)
    const float s1   = pt;
    const float num  = pp - s1 * s1;
    const float n2   = fmaxf(fmaf(s1 * s1, tt, fmaf(-2.0f * s1, s1, pp)), 0.0f);
    const float den  = fmaxf(sqrtf(n2), kEps);
    lsum += fmaxf(kMargin + num / den - s1, 0.0f);

    stage ^= 1;
  }

  if (lane == 0) wsum[wave] = lsum;
  __syncthreads();
  if (threadIdx.x == 0) {
    float b = 0.0f;
#pragma unroll
    for (int i = 0; i < kWaves; ++i) b += wsum[i];
    partial[blockIdx.x] = b;
  }
}

// Final reduction on one wave via the f32 WMMA path: with A = ones(16x4),
// D = A x B + C gives every row of D equal to the per-column sums of B, so
// accumulating 32 chunks of 64 partials and summing one accumulator VGPR
// across the wave yields 2x the grand total (lanes 0-15 hold row M=0,
// lanes 16-31 hold the identical row M=8). Deterministic, runs once.
__global__ __launch_bounds__(32) void triplet_loss_reduce_wmma(
    const float* __restrict__ partial, float* __restrict__ out, int n) {
  const int lane = threadIdx.x;  // 32 threads; EXEC all ones (WMMA requirement)

  v2f a; a.x = 1.0f; a.y = 1.0f;   // A = ones(16x4): 2 VGPRs per lane
  v8f c = {};                      // C/D = 16x16 f32 accumulator

  for (int base = 0; base < n; base += 64) {
    v2f b;                         // 64 distinct partials per WMMA step
    b.x = partial[base + lane * 2 + 0];
    b.y = partial[base + lane * 2 + 1];
    c = __builtin_amdgcn_wmma_f32_16x16x4_f32(
        /*neg_a=*/false, a, /*neg_b=*/false, b,
        /*c_mod=*/(short)0, c, /*reuse_a=*/false, /*reuse_b=*/false);
  }

  float v = c[0];
#pragma unroll
  for (int off = 16; off > 0; off >>= 1) v += __shfl_xor(v, off, 32);

  if (lane == 0) out[0] = v * 0.5f / (float)kN;   // undo M=0/M=8 duplication
}

extern "C" void kernel_launch(void* const* d_in, const int* in_sizes, int n_in,
                              void* d_out, int out_size, void* d_ws, size_t ws_size,
                              hipStream_t stream) {
  (void)in_sizes; (void)n_in; (void)out_size; (void)ws_size;
  const float* targets = (const float*)d_in[0];
  const float* preds   = (const float*)d_in[1];
  // d_in[2] = decode_lengths (int64) is unused by the reference computation.
  float* partial = (float*)d_ws;       // kBlocks floats = 8 KB
  float* out     = (float*)d_out;

  triplet_loss_main<<<kBlocks, kTPB, 0, stream>>>(targets, preds, partial);
  triplet_loss_reduce_wmma<<<1, 32, 0, stream>>>(partial, out, kBlocks);
}